// SignedSAGEConvolutionDeep_24352464568464
// MI455X (gfx1250) — compile-verified
//
#include <hip/hip_runtime.h>
#include <hip/hip_bf16.h>

typedef __attribute__((ext_vector_type(2))) float v2f;
typedef __attribute__((ext_vector_type(8))) float v8f;

#define D_FEAT 64

// ---------------------------------------------------------------------------
// Hardware f32 atomic add (lowers to global_atomic_add_f32; agg buffers fit in
// the 192MB L2, so these resolve at the L2 atomic units).
// ---------------------------------------------------------------------------
__device__ __forceinline__ void atomic_add_f32(float* p, float v) {
    __hip_atomic_fetch_add(p, v, __ATOMIC_RELAXED, __HIP_MEMORY_SCOPE_AGENT);
}

// ---------------------------------------------------------------------------
// Phase 0: init aggregation buffers with the self-loop contribution:
//   agg[i] = x[i], cnt[i] = 1
// ---------------------------------------------------------------------------
__global__ void sage_init_kernel(const float* __restrict__ x1,
                                 const float* __restrict__ x2,
                                 float* __restrict__ agg1,
                                 float* __restrict__ agg2,
                                 float* __restrict__ cnt1,
                                 float* __restrict__ cnt2,
                                 long total, int n) {
    long t = (long)blockIdx.x * blockDim.x + threadIdx.x;
    if (t < total) {
        agg1[t] = x1[t];
        agg2[t] = x2[t];
        if (t < n) { cnt1[t] = 1.0f; cnt2[t] = 1.0f; }
    }
}

// ---------------------------------------------------------------------------
// Phase 1: edge scatter.  thread -> (edge e, feature f).  The 64 threads of an
// edge read x[col*64+f] coalesced and atomically add to agg[row*64+f].
// ---------------------------------------------------------------------------
__global__ void sage_scatter_kernel(const float* __restrict__ x,
                                    const int* __restrict__ edges,  // [2][E]
                                    float* __restrict__ agg,
                                    float* __restrict__ cnt,
                                    int n_edges) {
    long t = (long)blockIdx.x * blockDim.x + threadIdx.x;
    long e = t >> 6;
    int  f = (int)(t & 63);
    if (e >= n_edges) return;
    int row = edges[e];
    int col = edges[(long)n_edges + e];
    atomic_add_f32(&agg[(long)row * D_FEAT + f], x[(long)col * D_FEAT + f]);
    if (f == 0) atomic_add_f32(&cnt[row], 1.0f);
}

// ---------------------------------------------------------------------------
// Phase 2: fused  [agg1/c1 | agg2/c2 | x1] @ W + bias, then row L2-normalize.
// One wave -> 16 rows x 64 cols via 4 x v_wmma_f32_16x16x4_f32 accumulators.
// W is staged in LDS pair-interleaved: ldsW[(kk*64+n)*2 + j] = W[2*kk+j][n],
// so a B fragment (lane(h,n) needs W[4k+2h][n],W[4k+2h+1][n]) is one b64 read
// at kk = 2k+h.  A fragment (lane(h,m) needs A[m][4k+2h..+1]) is one float2
// global load, scaled by 1/deg on the fly — the concat is never materialized.
// ---------------------------------------------------------------------------
__global__ __launch_bounds__(256) void sage_gemm_norm_kernel(
        const float* __restrict__ agg1, const float* __restrict__ agg2,
        const float* __restrict__ cnt1, const float* __restrict__ cnt2,
        const float* __restrict__ x1,   const float* __restrict__ W,
        const float* __restrict__ bias, float* __restrict__ out, int n) {
    __shared__ float ldsW[192 * 64];

    const int tid = threadIdx.x;
    // Cooperative, coalesced load of W into pair-interleaved LDS layout.
    for (int idx = tid; idx < 192 * 64; idx += 256) {
        const int row = idx >> 6;
        const int col = idx & 63;
        ldsW[((((row >> 1) << 6) | col) << 1) | (row & 1)] = W[idx];
    }
    __syncthreads();

    const int lane = tid & 31;
    const int wave = tid >> 5;
    const int m = lane & 15;   // M within tile (and N within a 16-wide tile)
    const int h = lane >> 4;   // lane half -> K sub-pair / row offset 8

    const long tile = (long)blockIdx.x * 8 + wave;
    const long row0 = tile << 4;
    long r = row0 + m;
    if (r > (long)(n - 1)) r = (long)(n - 1);   // clamp: keep EXEC all-ones

    const float* pa1 = agg1 + r * D_FEAT;
    const float* pa2 = agg2 + r * D_FEAT;
    const float* pax = x1   + r * D_FEAT;
    const float inv1 = 1.0f / fmaxf(cnt1[r], 1.0f);
    const float inv2 = 1.0f / fmaxf(cnt2[r], 1.0f);

    v8f acc0 = {}; v8f acc1 = {}; v8f acc2 = {}; v8f acc3 = {};
    const float* wbase = ldsW + (m << 1);

#pragma unroll
    for (int k = 0; k < 16; ++k) {              // cols 0..63 : agg1 / deg1
        const int c = (k << 2) + (h << 1);
        v2f a = *(const v2f*)(pa1 + c);
        a *= inv1;
        const float* wp = wbase + (((k << 1) + h) << 7);
        const v2f b0 = *(const v2f*)(wp);
        const v2f b1 = *(const v2f*)(wp + 32);
        const v2f b2 = *(const v2f*)(wp + 64);
        const v2f b3 = *(const v2f*)(wp + 96);
        acc0 = __builtin_amdgcn_wmma_f32_16x16x4_f32(false, a, false, b0, (short)0, acc0, false, false);
        acc1 = __builtin_amdgcn_wmma_f32_16x16x4_f32(false, a, false, b1, (short)0, acc1, false, false);
        acc2 = __builtin_amdgcn_wmma_f32_16x16x4_f32(false, a, false, b2, (short)0, acc2, false, false);
        acc3 = __builtin_amdgcn_wmma_f32_16x16x4_f32(false, a, false, b3, (short)0, acc3, false, false);
    }
#pragma unroll
    for (int k = 0; k < 16; ++k) {              // cols 64..127 : agg2 / deg2
        const int c = (k << 2) + (h << 1);
        v2f a = *(const v2f*)(pa2 + c);
        a *= inv2;
        const float* wp = wbase + ((32 + (k << 1) + h) << 7);
        const v2f b0 = *(const v2f*)(wp);
        const v2f b1 = *(const v2f*)(wp + 32);
        const v2f b2 = *(const v2f*)(wp + 64);
        const v2f b3 = *(const v2f*)(wp + 96);
        acc0 = __builtin_amdgcn_wmma_f32_16x16x4_f32(false, a, false, b0, (short)0, acc0, false, false);
        acc1 = __builtin_amdgcn_wmma_f32_16x16x4_f32(false, a, false, b1, (short)0, acc1, false, false);
        acc2 = __builtin_amdgcn_wmma_f32_16x16x4_f32(false, a, false, b2, (short)0, acc2, false, false);
        acc3 = __builtin_amdgcn_wmma_f32_16x16x4_f32(false, a, false, b3, (short)0, acc3, false, false);
    }
#pragma unroll
    for (int k = 0; k < 16; ++k) {              // cols 128..191 : x1
        const int c = (k << 2) + (h << 1);
        const v2f a = *(const v2f*)(pax + c);
        const float* wp = wbase + ((64 + (k << 1) + h) << 7);
        const v2f b0 = *(const v2f*)(wp);
        const v2f b1 = *(const v2f*)(wp + 32);
        const v2f b2 = *(const v2f*)(wp + 64);
        const v2f b3 = *(const v2f*)(wp + 96);
        acc0 = __builtin_amdgcn_wmma_f32_16x16x4_f32(false, a, false, b0, (short)0, acc0, false, false);
        acc1 = __builtin_amdgcn_wmma_f32_16x16x4_f32(false, a, false, b1, (short)0, acc1, false, false);
        acc2 = __builtin_amdgcn_wmma_f32_16x16x4_f32(false, a, false, b2, (short)0, acc2, false, false);
        acc3 = __builtin_amdgcn_wmma_f32_16x16x4_f32(false, a, false, b3, (short)0, acc3, false, false);
    }

    // Epilogue: bias, per-row L2 norm (row M=j+8h lives at VGPR j, lane-half h;
    // xor-shuffles 1/2/4/8 reduce within each 16-lane half), scaled store.
    const float bs0 = bias[m];
    const float bs1 = bias[16 + m];
    const float bs2 = bias[32 + m];
    const float bs3 = bias[48 + m];

#pragma unroll
    for (int j = 0; j < 8; ++j) {
        const float o0 = acc0[j] + bs0;
        const float o1 = acc1[j] + bs1;
        const float o2 = acc2[j] + bs2;
        const float o3 = acc3[j] + bs3;
        float s = o0 * o0 + o1 * o1 + o2 * o2 + o3 * o3;
        s += __shfl_xor(s, 1, 32);
        s += __shfl_xor(s, 2, 32);
        s += __shfl_xor(s, 4, 32);
        s += __shfl_xor(s, 8, 32);
        const float scale = 1.0f / fmaxf(sqrtf(s), 1e-12f);
        const long orow = row0 + j + (h << 3);
        if (orow < n) {
            float* po = out + orow * D_FEAT + m;
            po[0]  = o0 * scale;
            po[16] = o1 * scale;
            po[32] = o2 * scale;
            po[48] = o3 * scale;
        }
    }
}

// ---------------------------------------------------------------------------
extern "C" void kernel_launch(void* const* d_in, const int* in_sizes, int n_in,
                              void* d_out, int out_size, void* d_ws, size_t ws_size,
                              hipStream_t stream) {
    const float* x1    = (const float*)d_in[0];
    const float* x2    = (const float*)d_in[1];
    const int*   e_pos = (const int*)d_in[2];
    const int*   e_neg = (const int*)d_in[3];
    const float* W     = (const float*)d_in[4];
    const float* bias  = (const float*)d_in[5];
    float*       out   = (float*)d_out;

    const int n       = in_sizes[0] / D_FEAT;   // 100000
    const int n_edges = in_sizes[2] / 2;        // 1000000

    // Workspace layout: agg1[n*64] | agg2[n*64] | cnt1[n] | cnt2[n]
    float* agg1 = (float*)d_ws;
    float* agg2 = agg1 + (size_t)n * D_FEAT;
    float* cnt1 = agg2 + (size_t)n * D_FEAT;
    float* cnt2 = cnt1 + n;

    const long total = (long)n * D_FEAT;
    {
        const int blk = 256;
        const long grid = (total + blk - 1) / blk;
        sage_init_kernel<<<(unsigned)grid, blk, 0, stream>>>(x1, x2, agg1, agg2,
                                                             cnt1, cnt2, total, n);
    }
    {
        const int blk = 256;
        const long work = (long)n_edges * D_FEAT;
        const long grid = (work + blk - 1) / blk;
        sage_scatter_kernel<<<(unsigned)grid, blk, 0, stream>>>(x1, e_pos, agg1, cnt1, n_edges);
        sage_scatter_kernel<<<(unsigned)grid, blk, 0, stream>>>(x2, e_neg, agg2, cnt2, n_edges);
    }
    {
        const int tiles  = (n + 15) / 16;
        const int blocks = (tiles + 7) / 8;     // 8 waves (128 rows) per block
        sage_gemm_norm_kernel<<<blocks, 256, 0, stream>>>(agg1, agg2, cnt1, cnt2,
                                                          x1, W, bias, out, n);
    }
}